// LCNBlock_59356448030849
// MI455X (gfx1250) — compile-verified
//
#include <hip/hip_runtime.h>
#include <hip/hip_bf16.h>

typedef __attribute__((ext_vector_type(2))) float v2f;
typedef __attribute__((ext_vector_type(8))) float v8f;

#define J_    17
#define KC_   128
#define D_    2176          // 17 * 128
#define B_    8192
#define EPS_  1e-5f
#define LDSD  20            // padded LDS row stride (floats): conflict-free + 16B aligned

__device__ __forceinline__ float preluf(float v, float a) {
    return v >= 0.0f ? v : a * v;
}

// ---------------------------------------------------------------------------
// zero dense weight region (2 * D * D floats), float4-wide
__global__ __launch_bounds__(256)
void lcn_zero_kernel(float4* __restrict__ p, int n4) {
    int i = blockIdx.x * 256 + threadIdx.x;
    if (i < n4) p[i] = make_float4(0.f, 0.f, 0.f, 0.f);
}

// scatter sparse weights into both dense matrices
__global__ __launch_bounds__(256)
void lcn_scatter_kernel(const float* __restrict__ w1, const float* __restrict__ w2,
                        const int* __restrict__ rows, const int* __restrict__ cols,
                        float* __restrict__ W1d, float* __restrict__ W2d, int nnz) {
    int i = blockIdx.x * 256 + threadIdx.x;
    if (i < nnz) {
        size_t off = (size_t)rows[i] * D_ + (size_t)cols[i];
        W1d[off] = w1[i];
        W2d[off] = w2[i];
    }
}

// ---------------------------------------------------------------------------
// Tiled GEMM: Y[m,n] = sum_k A[m,k] * Wd[n,k] + bias[n]
// 128x128 block tile, BK=16, 8 wave32 waves (4 M x 2 N), WMMA f32 16x16x4.
// Optionally applies BN(scale/shift per joint) + PReLU to A elements on load.
// Emits deterministic per-block (sum, sumsq) partials: one joint per blockIdx.y.
template<bool TRANSFORM_A>
__global__ __launch_bounds__(256)
void lcn_gemm_kernel(const float* __restrict__ A,
                     const float* __restrict__ Wd,
                     const float* __restrict__ bias,
                     const float* __restrict__ sc,    // per-joint scale (TRANSFORM_A)
                     const float* __restrict__ sh,    // per-joint shift (TRANSFORM_A)
                     const float* __restrict__ pa,    // prelu alpha (TRANSFORM_A)
                     float* __restrict__ Y,
                     float* __restrict__ Pstats)      // [17][64][2]
{
    __shared__ float As[128][LDSD];
    __shared__ float Bs[128][LDSD];
    __shared__ float scJ[J_], shJ[J_];
    __shared__ float redS[8], redQ[8];

    const int tid   = threadIdx.x;
    const int lane  = tid & 31;
    const int wave  = tid >> 5;
    const int waveM = wave >> 1;       // 0..3
    const int waveN = wave & 1;        // 0..1
    const int l15   = lane & 15;
    const int h     = lane >> 4;       // 0 or 1
    const int m0    = blockIdx.x * 128;
    const int n0    = blockIdx.y * 128;

    float av = 0.0f;
    if (TRANSFORM_A) {
        if (tid < J_) { scJ[tid] = sc[tid]; shJ[tid] = sh[tid]; }
        av = pa[0];
    }
    __syncthreads();

    v8f acc[2][4];
    #pragma unroll
    for (int mi = 0; mi < 2; ++mi)
        #pragma unroll
        for (int ni = 0; ni < 4; ++ni)
            acc[mi][ni] = (v8f)0.0f;

    for (int kt = 0; kt < D_ / 16; ++kt) {
        const int kk = kt * 16;
        // stage A (128x16) and W (128x16) k-panels into LDS, float4-wide
        #pragma unroll
        for (int r = 0; r < 2; ++r) {
            int idx = tid + r * 256;            // 0..511
            int row = idx >> 2;                 // 0..127
            int c4  = (idx & 3) * 4;            // 0,4,8,12
            float4 va = *(const float4*)&A[(size_t)(m0 + row) * D_ + kk + c4];
            if (TRANSFORM_A) {
                int j = (kk + c4) >> 7;         // 4 consecutive cols share a joint
                float s = scJ[j], t = shJ[j];
                va.x = preluf(va.x * s + t, av);
                va.y = preluf(va.y * s + t, av);
                va.z = preluf(va.z * s + t, av);
                va.w = preluf(va.w * s + t, av);
            }
            *(float4*)&As[row][c4] = va;
            float4 vb = *(const float4*)&Wd[(size_t)(n0 + row) * D_ + kk + c4];
            *(float4*)&Bs[row][c4] = vb;
        }
        __syncthreads();

        #pragma unroll
        for (int kq = 0; kq < 4; ++kq) {
            const int kc = kq * 4 + 2 * h;
            // A frags: X[m = l15][k + 2h], X[m][k + 2h + 1]
            v2f a0 = *(const v2f*)&As[waveM * 32 +      l15][kc];
            v2f a1 = *(const v2f*)&As[waveM * 32 + 16 + l15][kc];
            // B frags: B[k][n] = W[n][k] -> same K-major pattern
            v2f bf[4];
            #pragma unroll
            for (int ni = 0; ni < 4; ++ni)
                bf[ni] = *(const v2f*)&Bs[waveN * 64 + ni * 16 + l15][kc];
            #pragma unroll
            for (int ni = 0; ni < 4; ++ni) {
                acc[0][ni] = __builtin_amdgcn_wmma_f32_16x16x4_f32(
                    false, a0, false, bf[ni], (short)0, acc[0][ni], false, false);
                acc[1][ni] = __builtin_amdgcn_wmma_f32_16x16x4_f32(
                    false, a1, false, bf[ni], (short)0, acc[1][ni], false, false);
            }
        }
        __syncthreads();
    }

    // epilogue: add bias, store Y, accumulate per-thread stats
    float s = 0.0f, sq = 0.0f;
    #pragma unroll
    for (int mi = 0; mi < 2; ++mi) {
        const int rBase = m0 + waveM * 32 + mi * 16 + 8 * h;
        #pragma unroll
        for (int ni = 0; ni < 4; ++ni) {
            const int col = n0 + waveN * 64 + ni * 16 + l15;
            const float bv = bias[col];
            #pragma unroll
            for (int e = 0; e < 8; ++e) {
                float v = acc[mi][ni][e] + bv;
                Y[(size_t)(rBase + e) * D_ + col] = v;
                s  += v;
                sq += v * v;
            }
        }
    }
    // wave reduction then block reduction (deterministic, no atomics)
    #pragma unroll
    for (int off = 16; off > 0; off >>= 1) {
        s  += __shfl_down(s,  off);
        sq += __shfl_down(sq, off);
    }
    if (lane == 0) { redS[wave] = s; redQ[wave] = sq; }
    __syncthreads();
    if (tid == 0) {
        float ts = 0.0f, tq = 0.0f;
        #pragma unroll
        for (int w = 0; w < 8; ++w) { ts += redS[w]; tq += redQ[w]; }
        size_t slot = ((size_t)blockIdx.y * 64 + blockIdx.x) * 2;
        Pstats[slot + 0] = ts;
        Pstats[slot + 1] = tq;
    }
}

// ---------------------------------------------------------------------------
// finalize BN stats: per joint, fold gamma/beta/mean/rsqrt(var+eps)
__global__ void lcn_finalize_kernel(const float* __restrict__ P,
                                    const float* __restrict__ g,
                                    const float* __restrict__ be,
                                    float* __restrict__ sc, float* __restrict__ sh) {
    int j = threadIdx.x;
    if (j >= J_) return;
    float s = 0.0f, sq = 0.0f;
    for (int m = 0; m < 64; ++m) {
        size_t slot = ((size_t)j * 64 + m) * 2;
        s  += P[slot + 0];
        sq += P[slot + 1];
    }
    const float inv_cnt = 1.0f / ((float)B_ * (float)KC_);
    float mean = s * inv_cnt;
    float var  = sq * inv_cnt - mean * mean;
    float istd = rsqrtf(var + EPS_);
    float scale = g[j] * istd;
    sc[j] = scale;
    sh[j] = be[j] - mean * scale;
}

// ---------------------------------------------------------------------------
// out = x + prelu(bn2(y2)); float4-wide, 4 consecutive cols share a joint
__global__ __launch_bounds__(256)
void lcn_final_kernel(const float* __restrict__ x, const float* __restrict__ y2,
                      const float* __restrict__ sc, const float* __restrict__ sh,
                      const float* __restrict__ pa, float* __restrict__ out) {
    size_t i = (size_t)blockIdx.x * 256 + threadIdx.x;   // float4 index
    size_t e = i * 4;
    int col = (int)(e % D_);
    int j = col >> 7;
    float s = sc[j], t = sh[j], a = pa[0];
    float4 y = *(const float4*)&y2[e];
    float4 xx = *(const float4*)&x[e];
    float4 o;
    o.x = xx.x + preluf(y.x * s + t, a);
    o.y = xx.y + preluf(y.y * s + t, a);
    o.z = xx.z + preluf(y.z * s + t, a);
    o.w = xx.w + preluf(y.w * s + t, a);
    *(float4*)&out[e] = o;
}

// ---------------------------------------------------------------------------
extern "C" void kernel_launch(void* const* d_in, const int* in_sizes, int n_in,
                              void* d_out, int out_size, void* d_ws, size_t ws_size,
                              hipStream_t stream) {
    (void)n_in; (void)out_size; (void)ws_size;
    const float* x    = (const float*)d_in[0];
    const float* w1   = (const float*)d_in[1];
    const float* b1   = (const float*)d_in[2];
    const float* g1   = (const float*)d_in[3];
    const float* be1  = (const float*)d_in[4];
    const float* w2   = (const float*)d_in[5];
    const float* b2   = (const float*)d_in[6];
    const float* g2   = (const float*)d_in[7];
    const float* be2  = (const float*)d_in[8];
    const float* pa   = (const float*)d_in[9];
    const int*   rows = (const int*)d_in[10];
    const int*   cols = (const int*)d_in[11];
    const int    nnz  = in_sizes[1];

    // workspace layout (floats)
    float* W1d = (float*)d_ws;                               // D*D
    float* W2d = W1d + (size_t)D_ * D_;                      // D*D
    float* Y1  = W2d + (size_t)D_ * D_;                      // B*D
    float* Y2  = Y1  + (size_t)B_ * D_;                      // B*D
    float* P1  = Y2  + (size_t)B_ * D_;                      // 17*64*2
    float* P2  = P1  + 17 * 64 * 2;                          // 17*64*2
    float* sc1 = P2  + 17 * 64 * 2;                          // 32
    float* sh1 = sc1 + 32;
    float* sc2 = sh1 + 32;
    float* sh2 = sc2 + 32;

    // 1. zero dense weights (both matrices contiguous)
    const int n4 = (2 * D_ * D_) / 4;                        // 2,367,488
    lcn_zero_kernel<<<(n4 + 255) / 256, 256, 0, stream>>>((float4*)W1d, n4);

    // 2. scatter sparse weights
    lcn_scatter_kernel<<<(nnz + 255) / 256, 256, 0, stream>>>(w1, w2, rows, cols,
                                                              W1d, W2d, nnz);

    dim3 gemm_grid(B_ / 128, D_ / 128);                      // (64, 17)

    // 3. y1 = x @ W1^T + b1, with stats partials
    lcn_gemm_kernel<false><<<gemm_grid, 256, 0, stream>>>(
        x, W1d, b1, nullptr, nullptr, nullptr, Y1, P1);

    // 4. BN1 stats -> scale/shift
    lcn_finalize_kernel<<<1, 32, 0, stream>>>(P1, g1, be1, sc1, sh1);

    // 5. y2 = prelu(bn1(y1)) @ W2^T + b2 (transform fused into A load), stats
    lcn_gemm_kernel<true><<<gemm_grid, 256, 0, stream>>>(
        Y1, W2d, b2, sc1, sh1, pa, Y2, P2);

    // 6. BN2 stats -> scale/shift
    lcn_finalize_kernel<<<1, 32, 0, stream>>>(P2, g2, be2, sc2, sh2);

    // 7. out = x + prelu(bn2(y2))
    const int nblk = (int)(((size_t)B_ * D_ / 4) / 256);     // 17408, exact
    lcn_final_kernel<<<nblk, 256, 0, stream>>>(x, Y2, sc2, sh2, pa, (float*)d_out);
}